// NemotronHTopkRouter_4174708212190
// MI455X (gfx1250) — compile-verified
//
#include <hip/hip_runtime.h>

#define HIDDEN 1536
#define NEXP   8
#define TOPK   2
#define WROW   1544   // 1536 + 8 floats pad -> expert rows 8 banks apart

typedef __attribute__((ext_vector_type(2))) float v2f;
typedef __attribute__((ext_vector_type(8))) float v8f;

__global__ __launch_bounds__(256)
void nemotron_topk_router_kernel(const float* __restrict__ hs,
                                 const float* __restrict__ w,
                                 const float* __restrict__ bias,
                                 int*   __restrict__ out_idx,
                                 float* __restrict__ out_w,
                                 int T)
{
    __shared__ float sW[NEXP * WROW];   // 48.25 KB padded weight
    __shared__ float sS[8][16][17];     // per-wave 16x16 logit tile (+pad)

    const int tid  = threadIdx.x;
    const int wave = tid >> 5;
    const int lane = tid & 31;

    // ---- stage router weight (8 x 1536 f32 = 3072 float4) into padded LDS ----
    {
        const float4* wv = (const float4*)w;
        #pragma unroll
        for (int i = 0; i < 12; ++i) {
            int g  = tid + i * 256;          // 0..3071, exact
            int r  = g / 384;                // expert row (384 float4 per row)
            int c4 = g - r * 384;
            float4 v = wv[g];
            *(float4*)&sW[r * WROW + c4 * 4] = v;
        }
    }
    __syncthreads();

    // ---- per-wave 16-token tile GEMM via V_WMMA_F32_16X16X4_F32 ----
    const int row   = lane & 15;             // token row within tile / M index
    const int hi    = lane >> 4;             // K-slot half select
    const int tile0 = blockIdx.x * 128 + wave * 16;
    int tclamp = tile0 + row; if (tclamp >= T) tclamp = T - 1;

    const float* arow = hs + (size_t)tclamp * HIDDEN + hi * 4;
    const float* brow = sW + (size_t)(row & 7) * WROW + hi * 4;

#if __has_builtin(__builtin_amdgcn_wmma_f32_16x16x4_f32)
    // Two accumulators: consecutive WMMAs are independent (no D->C RAW chain),
    // so they pipeline instead of stalling on VALU_DEP.
    v8f acc0 = {}, acc1 = {};
    for (int k0 = 0; k0 < HIDDEN; k0 += 8) {
        __builtin_prefetch(arow + k0 + 64, 0, 3);          // WGP-scope stream prefetch
        float4 a = *(const float4*)(arow + k0);            // global_load_b128
        float4 b = *(const float4*)(brow + k0);            // ds_load_b128
        v2f a01 = {a.x, a.y}, a23 = {a.z, a.w};
        v2f b01 = {b.x, b.y}, b23 = {b.z, b.w};
        // covers k0,k0+1 (low lanes) / k0+4,k0+5 (high lanes)
        acc0 = __builtin_amdgcn_wmma_f32_16x16x4_f32(false, a01, false, b01,
                                                     (short)0, acc0, false, false);
        // covers k0+2,k0+3 / k0+6,k0+7
        acc1 = __builtin_amdgcn_wmma_f32_16x16x4_f32(false, a23, false, b23,
                                                     (short)0, acc1, false, false);
    }
    v8f acc = acc0 + acc1;
    // D layout: VGPR r -> M = r + 8*hi, N = lane&15
    #pragma unroll
    for (int r = 0; r < 8; ++r)
        sS[wave][r + 8 * hi][lane & 15] = acc[r];
#else
    // Fallback (no f32 WMMA builtin): plain VALU dot products
    if (lane < 16) {
        float a8[NEXP] = {};
        const float* ap = hs + (size_t)tclamp * HIDDEN;
        for (int k = 0; k < HIDDEN; ++k) {
            float x = ap[k];
            #pragma unroll
            for (int e = 0; e < NEXP; ++e)
                a8[e] = fmaf(x, sW[e * WROW + k], a8[e]);
        }
        #pragma unroll
        for (int e = 0; e < NEXP; ++e) sS[wave][row][e] = a8[e];
    }
#endif
    __syncthreads();

    // ---- per-token sigmoid + top-2 + normalize (lanes 0..15, one token each) ----
    if (lane < 16) {
        int tok = tile0 + lane;
        if (tok < T) {
            float sc[NEXP], ch[NEXP];
            #pragma unroll
            for (int e = 0; e < NEXP; ++e) {
                float l = sS[wave][lane][e];
                float s = 1.0f / (1.0f + __expf(-l));
                sc[e] = s;
                ch[e] = s + bias[e];          // scores_for_choice
            }
            // top-1 (lowest index wins ties, matching lax.top_k stability)
            int i1 = 0; float b1 = ch[0];
            #pragma unroll
            for (int e = 1; e < NEXP; ++e)
                if (ch[e] > b1) { b1 = ch[e]; i1 = e; }
            // top-2 excluding i1
            int i2 = (i1 == 0) ? 1 : 0; float b2 = -3.402823466e38f;
            #pragma unroll
            for (int e = 0; e < NEXP; ++e)
                if (e != i1 && ch[e] > b2) { b2 = ch[e]; i2 = e; }

            float w1 = sc[i1], w2 = sc[i2];
            float inv = 1.0f / (w1 + w2 + 1e-20f);
            out_idx[tok * 2 + 0] = i1;
            out_idx[tok * 2 + 1] = i2;
            out_w [tok * 2 + 0] = w1 * inv;
            out_w [tok * 2 + 1] = w2 * inv;
        }
    }
}

extern "C" void kernel_launch(void* const* d_in, const int* in_sizes, int n_in,
                              void* d_out, int out_size, void* d_ws, size_t ws_size,
                              hipStream_t stream) {
    const float* hs   = (const float*)d_in[0];
    const float* w    = (const float*)d_in[1];
    const float* bias = (const float*)d_in[2];
    const int T = in_sizes[0] / HIDDEN;                 // 16384 tokens

    int*   out_idx = (int*)d_out;                       // first T*2 int32
    float* out_w   = (float*)d_out + (size_t)T * TOPK;  // then  T*2 f32

    const int grid = (T + 127) / 128;                   // 128 tokens per block
    nemotron_topk_router_kernel<<<grid, 256, 0, stream>>>(hs, w, bias,
                                                          out_idx, out_w, T);
}